// RGB_32134945309113
// MI455X (gfx1250) — compile-verified
//
#include <hip/hip_runtime.h>

#define N_SAMPLES      128
#define RAYS_PER_BLOCK 8
#define THREADS        (RAYS_PER_BLOCK * 32)
#define FAR_DIST       1e10f

// gfx1250 async global->LDS path (ASYNCcnt-tracked), guarded so both
// toolchains and the host pass still compile.
#if defined(__AMDGCN__) && defined(__gfx1250__) && \
    __has_builtin(__builtin_amdgcn_global_load_async_to_lds_b128)
#define USE_ASYNC_LDS 1
#else
#define USE_ASYNC_LDS 0
#endif

// gfx1250 f32 WMMA (16x16x4) used as an exact-f32 cross-lane reducer.
#if defined(__AMDGCN__) && defined(__gfx1250__) && \
    __has_builtin(__builtin_amdgcn_wmma_f32_16x16x4_f32)
#define USE_WMMA_REDUCE 1
#else
#define USE_WMMA_REDUCE 0
#endif

#if USE_ASYNC_LDS
typedef int v4i __attribute__((ext_vector_type(4)));
typedef __attribute__((address_space(1))) v4i* g_v4i_ptr;   // global
typedef __attribute__((address_space(3))) v4i* l_v4i_ptr;   // LDS
#endif
#if USE_WMMA_REDUCE
typedef float v2f __attribute__((ext_vector_type(2)));
typedef float v8f __attribute__((ext_vector_type(8)));
#endif

__global__ __launch_bounds__(THREADS) void nerf_rgb_kernel(
    const float* __restrict__ density,
    const float* __restrict__ color,
    const float* __restrict__ z_vals,
    const float* __restrict__ rays_d,
    float* __restrict__ out,
    int n_rays)
{
#if USE_ASYNC_LDS
    __shared__ __align__(16) float ctile[RAYS_PER_BLOCK * N_SAMPLES * 3]; // 12 KB
#endif
    const int lane = threadIdx.x & 31;
    const int wave = threadIdx.x >> 5;
    const int ray  = blockIdx.x * RAYS_PER_BLOCK + wave;   // wave-uniform
    if (ray >= n_rays) return;                             // whole-wave exit

    const float* __restrict__ csrc = color + (size_t)ray * (N_SAMPLES * 3);

#if USE_ASYNC_LDS
    // Stage this ray's color (384 floats) into LDS with coalesced async
    // 512B/wave transfers; overlaps with the scan math below.
    float* cdst = &ctile[wave * (N_SAMPLES * 3)];
#pragma unroll
    for (int j = 0; j < 3; ++j) {
        const int e = (j * 32 + lane) * 4;   // float index, 16B granules
        __builtin_amdgcn_global_load_async_to_lds_b128(
            (g_v4i_ptr)(const void*)(csrc + e),
            (l_v4i_ptr)(void*)(cdst + e),
            0, 0);
    }
#endif

    // ---- dists: z[s+1]-z[s], last padded with FAR ----
    const float4 zv = *(const float4*)(z_vals + (size_t)ray * N_SAMPLES + lane * 4);
    const float  zn = __shfl_down(zv.x, 1, 32);            // neighbor's first z
    const float d0 = zv.y - zv.x;
    const float d1 = zv.z - zv.y;
    const float d2 = zv.w - zv.z;
    const float d3 = (lane == 31) ? FAR_DIST : (zn - zv.w);

    // ---- ray direction norm (wave-uniform, L0-broadcast) ----
    const float rx = rays_d[(size_t)ray * 3 + 0];
    const float ry = rays_d[(size_t)ray * 3 + 1];
    const float rz = rays_d[(size_t)ray * 3 + 2];
    const float nrm = sqrtf(rx * rx + ry * ry + rz * rz);

    // ---- alpha / transmittance factors ----
    const float4 dn = *(const float4*)(density + (size_t)ray * N_SAMPLES + lane * 4);
    const float e0 = __expf(-fmaxf(dn.x, 0.f) * d0 * nrm); // = 1 - alpha0
    const float e1 = __expf(-fmaxf(dn.y, 0.f) * d1 * nrm);
    const float e2 = __expf(-fmaxf(dn.z, 0.f) * d2 * nrm);
    const float e3 = __expf(-fmaxf(dn.w, 0.f) * d3 * nrm);

    const float t0 = e0 + 1e-10f, t1 = e1 + 1e-10f;
    const float t2 = e2 + 1e-10f, t3 = e3 + 1e-10f;

    // local exclusive prefix products within the lane's 4 samples
    const float q1 = t0, q2 = t0 * t1, q3 = q2 * t2;
    const float P  = q3 * t3;                              // lane total

    // ---- 5-step wave32 multiplicative scan (Hillis-Steele) ----
    float v = P;
#pragma unroll
    for (int off = 1; off < 32; off <<= 1) {
        const float u = __shfl_up(v, off, 32);
        if (lane >= off) v *= u;
    }
    float E = __shfl_up(v, 1, 32);                         // exclusive prefix
    if (lane == 0) E = 1.0f;

    const float w0 = (1.f - e0) * E;
    const float w1 = (1.f - e1) * E * q1;
    const float w2 = (1.f - e2) * E * q2;
    const float w3 = (1.f - e3) * E * q3;

    // ---- weighted color accumulation (12 floats per lane) ----
    float4 ca, cb, cc;
#if USE_ASYNC_LDS
#if __has_builtin(__builtin_amdgcn_s_wait_asynccnt)
    __builtin_amdgcn_s_wait_asynccnt(0);
#else
    asm volatile("s_wait_asynccnt 0" ::: "memory");
#endif
    asm volatile("" ::: "memory");                         // order LDS reads after wait
    const float* cl = cdst + lane * 12;
    ca = *(const float4*)(cl + 0);
    cb = *(const float4*)(cl + 4);
    cc = *(const float4*)(cl + 8);
#else
    const float* cg = csrc + lane * 12;
    ca = *(const float4*)(cg + 0);
    cb = *(const float4*)(cg + 4);
    cc = *(const float4*)(cg + 8);
#endif
    // flat layout: [s0.r s0.g s0.b s1.r | s1.g s1.b s2.r s2.g | s2.b s3.r s3.g s3.b]
    float r = w0 * ca.x + w1 * ca.w + w2 * cb.z + w3 * cc.y;
    float g = w0 * ca.y + w1 * cb.x + w2 * cb.w + w3 * cc.z;
    float b = w0 * ca.z + w1 * cb.y + w2 * cc.x + w3 * cc.w;

#if USE_WMMA_REDUCE
    // Exact-f32 cross-lane reduction via V_WMMA_F32_16X16X4_F32.
    // A layout: VGPR0 = A[m][0] (lanes 0-15) / A[m][2] (lanes 16-31),
    //           VGPR1 = A[m][1]             / A[m][3].
    // => A = {r, g} per lane covers all 32 lanes in one 16x4 matrix.
    // B (4x16, same striping, roles swapped): column 0 selects K in {0,2} -> r,
    // column 1 selects K in {1,3} -> g. Second WMMA accumulates b into col 2.
    const int nl = lane & 15;
    v2f A1; A1.x = r; A1.y = g;
    v2f B1; B1.x = (nl == 0) ? 1.f : 0.f;  B1.y = (nl == 1) ? 1.f : 0.f;
    v2f A2; A2.x = b; A2.y = 0.f;
    v2f B2; B2.x = (nl == 2) ? 1.f : 0.f;  B2.y = 0.f;

    v8f acc = {};
    acc = __builtin_amdgcn_wmma_f32_16x16x4_f32(false, A1, false, B1,
                                                (short)0, acc, false, false);
    acc = __builtin_amdgcn_wmma_f32_16x16x4_f32(false, A2, false, B2,
                                                (short)0, acc, false, false);
    // lane n holds D[0..7][n], lane n+16 holds D[8..15][n]: sum rows locally,
    // then fold the two half-waves. Columns 0/1/2 carry r/g/b sums.
    float s = ((acc[0] + acc[1]) + (acc[2] + acc[3])) +
              ((acc[4] + acc[5]) + (acc[6] + acc[7]));
    s += __shfl_xor(s, 16, 32);
    if (lane < 3) out[(size_t)ray * 3 + lane] = s;
#else
    // ---- cross-lane sum (butterfly) fallback ----
#pragma unroll
    for (int off = 16; off > 0; off >>= 1) {
        r += __shfl_xor(r, off, 32);
        g += __shfl_xor(g, off, 32);
        b += __shfl_xor(b, off, 32);
    }
    if (lane == 0) {
        out[(size_t)ray * 3 + 0] = r;
        out[(size_t)ray * 3 + 1] = g;
        out[(size_t)ray * 3 + 2] = b;
    }
#endif
}

extern "C" void kernel_launch(void* const* d_in, const int* in_sizes, int n_in,
                              void* d_out, int out_size, void* d_ws, size_t ws_size,
                              hipStream_t stream) {
    (void)n_in; (void)out_size; (void)d_ws; (void)ws_size;
    const float* density = (const float*)d_in[0];   // [R, 128]
    const float* color   = (const float*)d_in[1];   // [R, 128, 3]
    const float* z_vals  = (const float*)d_in[2];   // [R, 128]
    const float* rays_d  = (const float*)d_in[3];   // [R, 3]
    float* out = (float*)d_out;                     // [R, 3]

    const int n_rays = in_sizes[0] / N_SAMPLES;
    const int blocks = (n_rays + RAYS_PER_BLOCK - 1) / RAYS_PER_BLOCK;
    nerf_rgb_kernel<<<blocks, THREADS, 0, stream>>>(density, color, z_vals,
                                                    rays_d, out, n_rays);
}